// EntREM_66460323938726
// MI455X (gfx1250) — compile-verified
//
#include <hip/hip_runtime.h>

// Problem constants (static in the reference: x_img [64,3,512,512] f32, pps=16, bins=64)
#define NB     64
#define NC     3
#define H      512
#define W      512
#define PPS    16
#define PH     32            // patch side in pixels
#define NBINS  64
#define PATCH_PIX (PH*PH)    // 1024
// side = round(sqrt(0.5*512*512)) = 362 -> //32*32 = 352, clamped -> 352
#define SIDE   352
#define HALF   176

typedef __attribute__((ext_vector_type(4))) unsigned int u32x4;
typedef __attribute__((ext_vector_type(8))) int          i32x8;
typedef __attribute__((ext_vector_type(4))) int          i32x4;
typedef __attribute__((ext_vector_type(2))) float        v2f;
typedef __attribute__((ext_vector_type(8))) float        v8f;

__device__ __forceinline__ int iclamp(int v, int lo, int hi) {
    return v < lo ? lo : (v > hi ? hi : v);
}

// ---------------------------------------------------------------------------
// Kernel 1: per-patch entropy. One block per patch. TDM stages the 32x32x3
// strided tile into LDS (one tensor_load_to_lds), then LDS histogram.
// Final 64-term entropy reduction done with one V_WMMA_F32_16X16X4_F32
// (A = 16x4 terms, B = ones -> D col0 holds the 16 row sums).
// ---------------------------------------------------------------------------
__global__ __launch_bounds__(256) void ent_kernel(const float* __restrict__ img,
                                                  float* __restrict__ ent) {
    // Single shared object -> guaranteed LDS offset 0 for the tile.
    __shared__ __align__(16) float smem[NC * PATCH_PIX + NBINS + 64];
    float*        tile = smem;                               // [3*1024] at LDS offset 0
    unsigned int* hist = (unsigned int*)(smem + NC * PATCH_PIX);
    float*        red  = smem + NC * PATCH_PIX + NBINS;

    const int gid = blockIdx.x;            // b*256 + pr*16 + pc
    const int b   = gid >> 8;
    const int pr  = (gid >> 4) & 15;
    const int pc  = gid & 15;
    const int t   = threadIdx.x;

    if (t < NBINS) hist[t] = 0u;

    if (t == 0) {
        // ---- Tensor DMA descriptor (D#): 3D tile 32(x) x 32(y) x 3(ch) ----
        unsigned long long gaddr =
            (unsigned long long)(const void*)(img + ((size_t)b * NC * H + (size_t)pr * PH) * W
                                                  + (size_t)pc * PH);
        u32x4 g0;
        g0[0] = 1u;                                          // count=1, user desc
        g0[1] = 0u;                                          // lds_addr = 0 (tile)
        g0[2] = (unsigned)(gaddr & 0xFFFFFFFFull);           // global_addr[31:0]
        g0[3] = (unsigned)((gaddr >> 32) & 0x01FFFFFFull)    // global_addr[56:32]
              | (2u << 30);                                  // type = 2 ("image")

        i32x8 g1;
        g1[0] = (2 << 16);                 // workgroup_mask=0 (not in cluster), data_size=4B
        g1[1] = (PH << 16);                // atomic_barrier_addr=0 | tensor_dim0[15:0]=32
        g1[2] = (PH << 16);                // tensor_dim0[31:16]=0 | tensor_dim1[15:0]=32
        g1[3] = (PH << 16);                // tensor_dim1[31:16]=0 | tile_dim0=32
        g1[4] = PH | (NC << 16);           // tile_dim1=32 | tile_dim2=3
        g1[5] = W;                         // tensor_dim0_stride[31:0] = 512
        g1[6] = ((H * W) & 0xFFFF) << 16;  // dim0_stride[47:32]=0 | dim1_stride[15:0]
        g1[7] = (H * W) >> 16;             // tensor_dim1_stride[47:16] (262144>>16 = 4)

        i32x4 g2;
        g2[0] = NC;                        // tensor_dim2 = 3
        g2[1] = 0;                         // tensor_dim3 (iterate disabled)
        g2[2] = 0;                         // tensor_dim2_stride lo (unused, tile_dim3=0)
        g2[3] = 0;                         // tensor_dim2_stride hi | tile_dim3=0
        i32x4 g3; g3[0] = 0; g3[1] = 0; g3[2] = 0; g3[3] = 0;

        // amdgpu-toolchain (clang-23) form: 6 args, extra int32x8 before cpol.
        i32x8 g4;
        g4[0] = 0; g4[1] = 0; g4[2] = 0; g4[3] = 0;
        g4[4] = 0; g4[5] = 0; g4[6] = 0; g4[7] = 0;

        __builtin_amdgcn_tensor_load_to_lds(g0, g1, g2, g3, g4, 0);
        __builtin_amdgcn_s_wait_tensorcnt(0);
    }
    __syncthreads();

    // Each thread handles 4 consecutive pixels of the 1024-pixel patch.
    float4 rr = ((const float4*)(tile))[t];
    float4 gg = ((const float4*)(tile + PATCH_PIX))[t];
    float4 bb = ((const float4*)(tile + 2 * PATCH_PIX))[t];

    float gr[4];
    gr[0] = 0.299f * rr.x + 0.587f * gg.x + 0.114f * bb.x;
    gr[1] = 0.299f * rr.y + 0.587f * gg.y + 0.114f * bb.y;
    gr[2] = 0.299f * rr.z + 0.587f * gg.z + 0.114f * bb.z;
    gr[3] = 0.299f * rr.w + 0.587f * gg.w + 0.114f * bb.w;
#pragma unroll
    for (int k = 0; k < 4; ++k) {
        int bin = iclamp((int)(gr[k] * (float)NBINS), 0, NBINS - 1); // trunc like astype(i32)
        atomicAdd(&hist[bin], 1u);
    }
    __syncthreads();

    if (t < NBINS) {
        float p = (float)hist[t] * (1.0f / (float)PATCH_PIX);
        red[t] = -p * logf(p + 1e-8f);
    }
    __syncthreads();

    // Wave 0 (full EXEC, wave-uniform guard) reduces the 64 terms with WMMA.
    if (t < 32) {
        const int l = t & 15;
        // 32-bit A-matrix 16x4 layout: VGPR0 = {lanes0-15: K=0, lanes16-31: K=2},
        //                              VGPR1 = {lanes0-15: K=1, lanes16-31: K=3}
        v2f A;
        if (t < 16) { A[0] = red[4 * l + 0]; A[1] = red[4 * l + 1]; }
        else        { A[0] = red[4 * l + 2]; A[1] = red[4 * l + 3]; }
        v2f Bones; Bones[0] = 1.0f; Bones[1] = 1.0f;
        v8f C = {};
        // D[m,n] = sum_k A[m,k] : row sums replicated across columns.
        v8f D = __builtin_amdgcn_wmma_f32_16x16x4_f32(
            false, A, false, Bones, (short)0, C, false, false);
        // Column 0: lanes 0-15 hold m=0..7 in VGPRs 0..7, lanes 16-31 hold m=8..15.
        float partial = D[0] + D[1] + D[2] + D[3] + D[4] + D[5] + D[6] + D[7];
        float other   = __shfl(partial, (t + 16) & 31, 32);
        if (t == 0) ent[gid] = partial + other;
    }
}

// ---------------------------------------------------------------------------
// Kernel 2: per-image argmax (first occurrence on ties) -> mask origin (y0,x0)
// ---------------------------------------------------------------------------
__global__ __launch_bounds__(256) void argmax_kernel(const float* __restrict__ ent,
                                                     int* __restrict__ yx) {
    __shared__ float sv[256];
    __shared__ int   si[256];
    const int b = blockIdx.x, t = threadIdx.x;
    sv[t] = ent[b * 256 + t];
    si[t] = t;
    for (int s = 128; s > 0; s >>= 1) {
        __syncthreads();
        if (t < s) {
            float v1 = sv[t], v2 = sv[t + s];
            int   i1 = si[t], i2 = si[t + s];
            if (v2 > v1 || (v2 == v1 && i2 < i1)) { sv[t] = v2; si[t] = i2; }
        }
    }
    if (t == 0) {
        int p  = si[0];
        int cy = (p >> 4) * PH + PH / 2;   // (pr+0.5)*32
        int cx = (p & 15) * PH + PH / 2;
        yx[2 * b]     = iclamp(cy - HALF, 0, H - SIDE);
        yx[2 * b + 1] = iclamp(cx - HALF, 0, W - SIDE);
    }
}

// ---------------------------------------------------------------------------
// Kernel 3: streaming masked copy, float4 (B128) loads/stores. Bandwidth pass.
// ---------------------------------------------------------------------------
__global__ __launch_bounds__(256) void mask_kernel(const float* __restrict__ img,
                                                   const int* __restrict__ yx,
                                                   float* __restrict__ out) {
    const int gid = blockIdx.x;            // 768 blocks per image (256 per channel)
    const int b   = gid / 768;
    const int rem = gid - b * 768;
    const int c   = rem >> 8;
    const int v   = ((rem & 255) << 8) + (int)threadIdx.x;  // vec index in channel, 0..65535
    const int y   = v >> 7;
    const int x   = (v & 127) << 2;

    const int y0 = yx[2 * b];
    const int x0 = yx[2 * b + 1];

    const size_t off = ((size_t)(b * NC + c) * H + (size_t)y) * W + (size_t)x;
    float4 val = *(const float4*)(img + off);

    if (y >= y0 && y < y0 + SIDE) {
        if (x + 0 >= x0 && x + 0 < x0 + SIDE) val.x = 0.0f;
        if (x + 1 >= x0 && x + 1 < x0 + SIDE) val.y = 0.0f;
        if (x + 2 >= x0 && x + 2 < x0 + SIDE) val.z = 0.0f;
        if (x + 3 >= x0 && x + 3 < x0 + SIDE) val.w = 0.0f;
    }
    *(float4*)(out + off) = val;
}

// ---------------------------------------------------------------------------
extern "C" void kernel_launch(void* const* d_in, const int* in_sizes, int n_in,
                              void* d_out, int out_size, void* d_ws, size_t ws_size,
                              hipStream_t stream) {
    const float* img = (const float*)d_in[0];
    float* ent = (float*)d_ws;                                   // 64*256 floats
    int*   yx  = (int*)((char*)d_ws + NB * 256 * sizeof(float)); // 64*2 ints
    float* out = (float*)d_out;

    ent_kernel   <<<NB * PPS * PPS, 256, 0, stream>>>(img, ent);
    argmax_kernel<<<NB,             256, 0, stream>>>(ent, yx);
    mask_kernel  <<<NB * 768,       256, 0, stream>>>(img, yx, out);
}